// Multiscale_MINE_oldV_68642167324939
// MI455X (gfx1250) — compile-verified
//
#include <hip/hip_runtime.h>
#include <hip/hip_bf16.h>

// ---------------- model dims ----------------
#define NBANDS 12
#define WFFT   4096
#define HDIM   256
#define HC2    512
#define HC4    1024
#define KTS    8
#define SPL    16
#define HOPN   2048

typedef __attribute__((ext_vector_type(8))) float v8f;
typedef __attribute__((ext_vector_type(2))) float v2f;

// =====================================================================
// 1) Band split: windowed FFT -> per-band mask -> IFFT -> * synthesis win
//    One block per band; 4096-pt radix-2 FFT fully in LDS (32 KB).
// =====================================================================
__global__ __launch_bounds__(256) void fft_band_kernel(
    const float* __restrict__ audio, const float* __restrict__ awin,
    const float* __restrict__ swin,  const float* __restrict__ masks,
    float* __restrict__ curr)
{
    __shared__ float re[WFFT];
    __shared__ float im[WFFT];
    const int b = blockIdx.x, t = threadIdx.x;

    // bit-reversed windowed load
    for (int i = t; i < WFFT; i += 256) {
        int r = (int)(__brev((unsigned)i) >> 20);   // 12-bit reverse
        re[r] = audio[i] * awin[i];
        im[r] = 0.f;
    }
    __syncthreads();

    // forward FFT (DIT)
    for (int s = 1; s <= 12; ++s) {
        const int half = 1 << (s - 1);
        for (int x = t; x < WFFT / 2; x += 256) {
            int grp = x >> (s - 1);
            int pos = x & (half - 1);
            int i0 = (grp << s) | pos;
            int i1 = i0 + half;
            float ang = -6.283185307179586f * (float)pos / (float)(1 << s);
            float sn, cs; sincosf(ang, &sn, &cs);
            float xr = re[i1], xi = im[i1];
            float tr = xr * cs - xi * sn;
            float ti = xr * sn + xi * cs;
            float ur = re[i0], ui = im[i0];
            re[i0] = ur + tr; im[i0] = ui + ti;
            re[i1] = ur - tr; im[i1] = ui - ti;
        }
        __syncthreads();
    }

    // apply band mask (natural order)
    for (int i = t; i < WFFT; i += 256) {
        float m = masks[(long)b * WFFT + i];
        re[i] *= m; im[i] *= m;
    }
    __syncthreads();

    // bit-reverse permutation in place (pairs disjoint)
    for (int i = t; i < WFFT; i += 256) {
        int r = (int)(__brev((unsigned)i) >> 20);
        if (i < r) {
            float a0 = re[i], a1 = im[i];
            re[i] = re[r]; im[i] = im[r];
            re[r] = a0;    im[r] = a1;
        }
    }
    __syncthreads();

    // inverse FFT (conjugate twiddle), scale by 1/N at the end
    for (int s = 1; s <= 12; ++s) {
        const int half = 1 << (s - 1);
        for (int x = t; x < WFFT / 2; x += 256) {
            int grp = x >> (s - 1);
            int pos = x & (half - 1);
            int i0 = (grp << s) | pos;
            int i1 = i0 + half;
            float ang = 6.283185307179586f * (float)pos / (float)(1 << s);
            float sn, cs; sincosf(ang, &sn, &cs);
            float xr = re[i1], xi = im[i1];
            float tr = xr * cs - xi * sn;
            float ti = xr * sn + xi * cs;
            float ur = re[i0], ui = im[i0];
            re[i0] = ur + tr; im[i0] = ui + ti;
            re[i1] = ur - tr; im[i1] = ui - ti;
        }
        __syncthreads();
    }

    const float scale = 1.0f / (float)WFFT;
    for (int i = t; i < WFFT; i += 256)
        curr[(long)b * WFFT + i] = re[i] * scale * swin[i];
}

// =====================================================================
// 2) Spiral net: 16-layer 512x512 ELU MLP, one block per pose (2 blocks)
// =====================================================================
__global__ __launch_bounds__(512) void spiral_kernel(
    const float* __restrict__ poseJ, const float* __restrict__ poseM,
    const float* __restrict__ sw, const float* __restrict__ sb,
    float* __restrict__ spout)
{
    __shared__ float xb[HC2];
    __shared__ float yb[HC2];
    const int t = threadIdx.x;
    const float* pose = (blockIdx.x == 0) ? poseJ : poseM;
    xb[t] = pose[t];
    __syncthreads();
    for (int i = 0; i < SPL; ++i) {
        const float* w = sw + ((long)i * HC2 + t) * HC2;
        float s = sb[(long)i * HC2 + t];
        for (int j = 0; j < HC2; j += 4)
            s += w[j] * xb[j] + w[j+1] * xb[j+1] + w[j+2] * xb[j+2] + w[j+3] * xb[j+3];
        if (i < SPL - 1) s = (s > 0.f) ? s : expm1f(s);   // ELU
        yb[t] = s;
        __syncthreads();
        xb[t] = yb[t];
        __syncthreads();
    }
    spout[(long)blockIdx.x * HC2 + t] = xb[t];
}

// =====================================================================
// 3) LayerNorm over the band spectra (12 x 4096), builds iso on the fly
// =====================================================================
__global__ __launch_bounds__(256) void ln_bands_kernel(
    const float* __restrict__ curr, int shift,
    const float* __restrict__ gw, const float* __restrict__ gb,
    float* __restrict__ out)
{
    __shared__ float xb[WFFT];
    __shared__ float red[256];
    const int f = blockIdx.x, t = threadIdx.x;
    const float* c = curr + (long)f * WFFT;
    float s = 0.f;
    for (int i = t; i < WFFT; i += 256) {
        float v = c[i];
        if (shift && i < HOPN) v += c[i + HOPN];
        xb[i] = v; s += v;
    }
    red[t] = s; __syncthreads();
    for (int o = 128; o > 0; o >>= 1) { if (t < o) red[t] += red[t + o]; __syncthreads(); }
    const float mu = red[0] * (1.f / WFFT);
    __syncthreads();
    float q = 0.f;
    for (int i = t; i < WFFT; i += 256) { float d = xb[i] - mu; q += d * d; }
    red[t] = q; __syncthreads();
    for (int o = 128; o > 0; o >>= 1) { if (t < o) red[t] += red[t + o]; __syncthreads(); }
    const float rstd = rsqrtf(red[0] * (1.f / WFFT) + 1e-5f);
    for (int i = t; i < WFFT; i += 256)
        out[(long)f * WFFT + i] = (xb[i] - mu) * rstd * gw[i] + gb[i];
}

// =====================================================================
// 4) LayerNorm of concat(out[f] (512), sp (512)) with per-(k,f) affine
// =====================================================================
__global__ __launch_bounds__(256) void ln_cat_kernel(
    const float* __restrict__ outv, const float* __restrict__ sp,
    const float* __restrict__ gw, const float* __restrict__ gb,
    float* __restrict__ rin)
{
    __shared__ float xb[HC4];
    __shared__ float red[256];
    const int f = blockIdx.x, t = threadIdx.x;
    float s = 0.f;
    for (int i = t; i < HC4; i += 256) {
        float v = (i < HC2) ? outv[(long)f * HC2 + i] : sp[i - HC2];
        xb[i] = v; s += v;
    }
    red[t] = s; __syncthreads();
    for (int o = 128; o > 0; o >>= 1) { if (t < o) red[t] += red[t + o]; __syncthreads(); }
    const float mu = red[0] * (1.f / HC4);
    __syncthreads();
    float q = 0.f;
    for (int i = t; i < HC4; i += 256) { float d = xb[i] - mu; q += d * d; }
    red[t] = q; __syncthreads();
    for (int o = 128; o > 0; o >>= 1) { if (t < o) red[t] += red[t + o]; __syncthreads(); }
    const float rstd = rsqrtf(red[0] * (1.f / HC4) + 1e-5f);
    for (int i = t; i < HC4; i += 256)
        rin[(long)f * HC4 + i] =
            (xb[i] - mu) * rstd * gw[(long)f * HC4 + i] + gb[(long)f * HC4 + i];
}

// =====================================================================
// 5) WMMA RNN cell:
//    grid = (2*NBANDS, 16 row-tiles) = 384 blocks per matvec batch.
//    Each block: one 16-row tile; 8 waves split K 8 ways, each wave runs
//    two interleaved V_WMMA_F32_16X16X4_F32 chains on its K-slice
//    (flat loop, unroll 4 -> clause of 8 global_load_b64 + 8 wmma),
//    then cross-wave LDS reduction + bias + ReLU epilogue.
//    h = relu( Wih[f,d] @ x[f] + Whh[f,d] @ hprev[f,d] + bi + bh )
// =====================================================================
__global__ __launch_bounds__(256) void gemv_rnn_kernel(
    const float* __restrict__ Wm, long wsF, long wsD, int K,
    const float* __restrict__ X,  long xsF,
    const float* __restrict__ Wh, long whsF, long whsD,
    const float* __restrict__ Hv, long hsF,  long hsD,
    const float* __restrict__ Bi, long bisF, long bisD,
    const float* __restrict__ Bh, long bhsF, long bhsD,
    float* __restrict__ Hout, long hosF, long hosD,
    float* __restrict__ Ocat, long ocsF)
{
    __shared__ float xs[WFFT];
    __shared__ float hsh[HDIM];
    __shared__ float red[8][16];
    const int f  = blockIdx.x >> 1;
    const int d  = blockIdx.x & 1;
    const int rt = blockIdx.y << 4;        // 16-row tile base
    const int tid = threadIdx.x;

    const float* xp = X + (long)f * xsF;
    for (int i = tid; i < K; i += 256) xs[i] = xp[i];
    if (Wh) hsh[tid] = Hv[(long)f * hsF + (long)d * hsD + tid];
    __syncthreads();

    const int lane = tid & 31;
    const int wave = tid >> 5;
    const int mlo  = lane & 15;
    const int kh   = (lane >> 4) << 1;     // 0 (lanes 0-15) or 2 (lanes 16-31)

    // per-wave contiguous K-slice, two interleaved accumulator chains
    const float* wrow = Wm + (long)f * wsF + (long)d * wsD + (long)(rt + mlo) * K
                        + (long)wave * (K >> 3);
    const float* xw = xs + wave * (K >> 3);
    const int Kw = K >> 3;
    v8f acc0 = {}; v8f acc1 = {};

    #pragma unroll 4
    for (int k = 0; k < Kw; k += 8) {
        if ((k & 255) == 0)
            __builtin_prefetch(wrow + k + 1024 + kh, 0, 0);   // stream-ahead -> GL2
        v2f b0; b0.x = xw[k + kh];     b0.y = xw[k + kh + 1];
        v2f b1; b1.x = xw[k + 4 + kh]; b1.y = xw[k + 4 + kh + 1];
        v2f a0 = *(const v2f*)(wrow + k + kh);
        v2f a1 = *(const v2f*)(wrow + k + 4 + kh);
        acc0 = __builtin_amdgcn_wmma_f32_16x16x4_f32(false, a0, false, b0, (short)0, acc0, false, false);
        acc1 = __builtin_amdgcn_wmma_f32_16x16x4_f32(false, a1, false, b1, (short)0, acc1, false, false);
    }
    if (Wh) {   // hidden-state matvec (256x256): per-wave 32-element slice
        const float* urow = Wh + (long)f * whsF + (long)d * whsD + (long)(rt + mlo) * HDIM
                            + (long)wave * 32;
        const float* hw = hsh + wave * 32;
        #pragma unroll
        for (int k = 0; k < 32; k += 8) {
            v2f b0; b0.x = hw[k + kh];     b0.y = hw[k + kh + 1];
            v2f b1; b1.x = hw[k + 4 + kh]; b1.y = hw[k + 4 + kh + 1];
            v2f a0 = *(const v2f*)(urow + k + kh);
            v2f a1 = *(const v2f*)(urow + k + 4 + kh);
            acc0 = __builtin_amdgcn_wmma_f32_16x16x4_f32(false, a0, false, b0, (short)0, acc0, false, false);
            acc1 = __builtin_amdgcn_wmma_f32_16x16x4_f32(false, a1, false, b1, (short)0, acc1, false, false);
        }
    }
    acc0 = acc0 + acc1;

    // D layout: lane 0 rows rt..rt+7 in acc0[0..7]; lane 16 rows rt+8..rt+15
    if ((lane & 15) == 0) {
        const int hi = (lane >> 4) << 3;   // 0 or 8
        for (int v = 0; v < 8; ++v) red[wave][hi + v] = acc0[v];
    }
    __syncthreads();

    if (tid < 16) {
        float s = 0.f;
        for (int w = 0; w < 8; ++w) s += red[w][tid];
        const int row = rt + tid;
        float val = s + Bi[(long)f * bisF + (long)d * bisD + row]
                      + Bh[(long)f * bhsF + (long)d * bhsD + row];
        val = val > 0.f ? val : 0.f;
        Hout[(long)f * hosF + (long)d * hosD + row] = val;
        if (Ocat) Ocat[(long)f * ocsF + (long)d * HDIM + row] = val;
    }
}

// =====================================================================
// 6) FFN head: o[f] = elu( dot(out[f], ffn_w[k,f]) + ffn_b[k,f] )
// =====================================================================
__global__ __launch_bounds__(256) void ffn_kernel(
    const float* __restrict__ outv, const float* __restrict__ fw,
    const float* __restrict__ fb, float* __restrict__ o)
{
    __shared__ float red[256];
    const int f = blockIdx.x, t = threadIdx.x;
    const float* a = outv + (long)f * HC2;
    const float* w = fw   + (long)f * HC2;
    float s = 0.f;
    for (int i = t; i < HC2; i += 256) s += a[i] * w[i];
    red[t] = s; __syncthreads();
    for (int off = 128; off > 0; off >>= 1) { if (t < off) red[t] += red[t + off]; __syncthreads(); }
    if (t == 0) {
        float v = red[0] + fb[f];
        o[f] = (v > 0.f) ? v : expm1f(v);
    }
}

// =====================================================================
// 7) Finalize: mine = joint - log(exp(marg));  out = [-sum(mine), mine]
// =====================================================================
__global__ __launch_bounds__(128) void finalize_kernel(
    const float* __restrict__ joint, const float* __restrict__ marg,
    float* __restrict__ out)
{
    __shared__ float s[(KTS - 1) * NBANDS];
    const int t = threadIdx.x;
    if (t < (KTS - 1) * NBANDS) {
        float m = joint[t] - logf(expf(marg[t]));
        s[t] = m;
        out[1 + t] = m;
    }
    __syncthreads();
    if (t == 0) {
        float tot = 0.f;
        for (int i = 0; i < (KTS - 1) * NBANDS; ++i) tot += s[i];
        out[0] = -tot;
    }
}

// =====================================================================
// Host orchestration
// =====================================================================
extern "C" void kernel_launch(void* const* d_in, const int* in_sizes, int n_in,
                              void* d_out, int out_size, void* d_ws, size_t ws_size,
                              hipStream_t stream) {
    const float* audio   = (const float*)d_in[0];
    const float* poseJ   = (const float*)d_in[1];
    const float* poseM   = (const float*)d_in[2];
    const float* awin    = (const float*)d_in[3];
    const float* swin    = (const float*)d_in[4];
    const float* masks   = (const float*)d_in[5];
    const float* fln_w   = (const float*)d_in[6];
    const float* fln_b   = (const float*)d_in[7];
    const float* rf_w_ih0 = (const float*)d_in[8];
    const float* rf_b_ih0 = (const float*)d_in[10];
    const float* rf_b_hh0 = (const float*)d_in[11];
    const float* rf_w_ihL = (const float*)d_in[12];
    const float* rf_b_ihL = (const float*)d_in[14];
    const float* rf_b_hhL = (const float*)d_in[15];
    const float* rl_ln_w  = (const float*)d_in[16];
    const float* rl_ln_b  = (const float*)d_in[17];
    const float* rl_w_ih0 = (const float*)d_in[18];
    const float* rl_w_hh0 = (const float*)d_in[19];
    const float* rl_b_ih0 = (const float*)d_in[20];
    const float* rl_b_hh0 = (const float*)d_in[21];
    const float* rl_w_ihL = (const float*)d_in[22];
    const float* rl_w_hhL = (const float*)d_in[23];
    const float* rl_b_ihL = (const float*)d_in[24];
    const float* rl_b_bhL = (const float*)d_in[25];
    const float* ffn_w    = (const float*)d_in[26];
    const float* ffn_b    = (const float*)d_in[27];
    const float* sp_w     = (const float*)d_in[28];
    const float* sp_b     = (const float*)d_in[29];

    float* ws = (float*)d_ws;
    float* curr = ws;                         // 12*4096
    float* xln  = curr + NBANDS * WFFT;       // 12*4096
    float* sp   = xln  + NBANDS * WFFT;       // 2*512
    float* outA = sp   + 2 * HC2;             // 12*512
    float* outB = outA + NBANDS * HC2;        // 12*512
    float* rin  = outB + NBANDS * HC2;        // 12*1024
    float* stA  = rin  + NBANDS * HC4;        // 12*8*256
    float* stB  = stA  + NBANDS * 8 * HDIM;   // 12*8*256
    float* jm   = stB  + NBANDS * 8 * HDIM;   // 2*84

    const dim3 ggrid(2 * NBANDS, HDIM / 16);  // (f,d) x 16 row-tiles

    // Band split FFT (curr identical for both frames; iso2 adds tail shift)
    fft_band_kernel<<<NBANDS, 256, 0, stream>>>(audio, awin, swin, masks, curr);
    spiral_kernel<<<2, 512, 0, stream>>>(poseJ, poseM, sp_w, sp_b, sp);

    for (int pass = 0; pass < 2; ++pass) {
        const float* spv = sp + (long)pass * HC2;
        float* outsbuf = jm + (long)pass * (KTS - 1) * NBANDS;

        ln_bands_kernel<<<NBANDS, 256, 0, stream>>>(curr, pass, fln_w, fln_b, xln);

        // ---- rf rnn4, h0 == 0 (hidden matvec skipped; both biases kept) ----
        gemv_rnn_kernel<<<ggrid, 256, 0, stream>>>(
            rf_w_ih0, 2L * HDIM * WFFT, (long)HDIM * WFFT, WFFT,
            xln, (long)WFFT,
            nullptr, 0, 0, nullptr, 0, 0,
            rf_b_ih0, 2L * HDIM, (long)HDIM,
            rf_b_hh0, 2L * HDIM, (long)HDIM,
            stA, 8L * HDIM, (long)HDIM,                 // st slots 0,1
            outA, (long)HC2);
        float* po = outA; float* pn = outB;
        for (int l = 0; l < 3; ++l) {
            gemv_rnn_kernel<<<ggrid, 256, 0, stream>>>(
                rf_w_ihL + (long)l * 2 * HDIM * HC2, 6L * HDIM * HC2, (long)HDIM * HC2, HC2,
                po, (long)HC2,
                nullptr, 0, 0, nullptr, 0, 0,
                rf_b_ihL + (long)l * 2 * HDIM, 6L * HDIM, (long)HDIM,
                rf_b_hhL + (long)l * 2 * HDIM, 6L * HDIM, (long)HDIM,
                stA + (long)(2 * l + 2) * HDIM, 8L * HDIM, (long)HDIM,
                pn, (long)HC2);
            float* tmp = po; po = pn; pn = tmp;
        }

        float* st = stA; float* stn = stB;
        for (int k = 0; k < KTS - 1; ++k) {
            // FFN uses pre-update `out`
            ffn_kernel<<<NBANDS, 256, 0, stream>>>(
                po, ffn_w + (long)k * (NBANDS + 1) * HC2, ffn_b + (long)k * (NBANDS + 1),
                outsbuf + (long)k * NBANDS);
            // rin = LN(concat(out, sp))
            ln_cat_kernel<<<NBANDS, 256, 0, stream>>>(
                po, spv, rl_ln_w + (long)k * NBANDS * HC4, rl_ln_b + (long)k * NBANDS * HC4, rin);

            // rl layer 0: K=1024 input matvec + 256x256 hidden matvec
            gemv_rnn_kernel<<<ggrid, 256, 0, stream>>>(
                rl_w_ih0 + (long)k * NBANDS * 2 * HDIM * HC4, 2L * HDIM * HC4, (long)HDIM * HC4, HC4,
                rin, (long)HC4,
                rl_w_hh0 + (long)k * NBANDS * 2 * HDIM * HDIM, 2L * HDIM * HDIM, (long)HDIM * HDIM,
                st, 8L * HDIM, (long)HDIM,
                rl_b_ih0 + (long)k * NBANDS * 2 * HDIM, 2L * HDIM, (long)HDIM,
                rl_b_hh0 + (long)k * NBANDS * 2 * HDIM, 2L * HDIM, (long)HDIM,
                stn, 8L * HDIM, (long)HDIM,
                pn, (long)HC2);

            float* qo = pn; float* qn = po;
            for (int l = 0; l < 3; ++l) {
                gemv_rnn_kernel<<<ggrid, 256, 0, stream>>>(
                    rl_w_ihL + ((long)k * NBANDS * 6 + l * 2) * HDIM * HC2, 6L * HDIM * HC2, (long)HDIM * HC2, HC2,
                    qo, (long)HC2,
                    rl_w_hhL + ((long)k * NBANDS * 6 + l * 2) * HDIM * HDIM, 6L * HDIM * HDIM, (long)HDIM * HDIM,
                    st + (long)(2 * l + 2) * HDIM, 8L * HDIM, (long)HDIM,
                    rl_b_ihL + ((long)k * NBANDS * 6 + l * 2) * HDIM, 6L * HDIM, (long)HDIM,
                    rl_b_bhL + ((long)k * NBANDS * 6 + l * 2) * HDIM, 6L * HDIM, (long)HDIM,
                    stn + (long)(2 * l + 2) * HDIM, 8L * HDIM, (long)HDIM,
                    qn, (long)HC2);
                float* tmp = qo; qo = qn; qn = tmp;
            }
            po = qo; pn = qn;
            float* tmp = st; st = stn; stn = tmp;
        }
    }

    finalize_kernel<<<1, 128, 0, stream>>>(jm, jm + (KTS - 1) * NBANDS, (float*)d_out);
}